// CopyModule_72988674228842
// MI455X (gfx1250) — compile-verified
//
#include <hip/hip_runtime.h>
#include <math.h>

#define BB 4
#define TT 256
#define TS 512
#define HH 1024
#define VV 32100
#define CHUNK 8192

typedef float v2f __attribute__((ext_vector_type(2)));
typedef float v8f __attribute__((ext_vector_type(8)));

// ---------------------------------------------------------------------------
// Kernel 1: encW[b*TS+s] = dot(enc[b,s,:], W_gen[0:H])   (2048 rows, trivial)
// One wave (32 lanes) per row, float4 vectorized, shuffle reduction.
// ---------------------------------------------------------------------------
__global__ void k_encw(const float* __restrict__ enc,
                       const float* __restrict__ wgen,
                       float* __restrict__ encW) {
    int wave = threadIdx.x >> 5;
    int lane = threadIdx.x & 31;
    int row  = blockIdx.x * 8 + wave;                 // 0..2047
    const float4* p = (const float4*)(enc + (size_t)row * HH);
    const float4* w = (const float4*)wgen;
    float acc = 0.f;
    for (int i = lane; i < HH / 4; i += 32) {
        float4 a = p[i];
        float4 b = w[i];
        acc += a.x * b.x + a.y * b.y + a.z * b.z + a.w * b.w;
    }
    for (int off = 16; off > 0; off >>= 1)
        acc += __shfl_down(acc, off, 32);
    if (lane == 0) encW[row] = acc;
}

// ---------------------------------------------------------------------------
// Kernel 2: p_gen via V_WMMA_F32_16X16X4_F32.
//   p_gen_logit[b,t] = attn[b,t,:].encW[b,:] + dec[b,t,:].Wgen[H:2H] + b_gen
// One wave per 16 target rows. A = 16x4 tile of attn (then dec) rows.
// B = 4x16 tile: column 0 carries the vector, columns 1..15 are zero.
// Same f32 accumulator across both K loops (S=512, then H=1024).
// A layout: lanes 0-15 -> M, K in {0,1}; lanes 16-31 -> M, K in {2,3}.
// B layout mirrors: lanes 0-15 -> N, K in {0,1}; lanes 16-31 -> N, K {2,3}.
// C layout: VGPR r -> M=r (lanes 0-15, N=lane) / M=8+r (lanes 16-31).
// ---------------------------------------------------------------------------
__global__ void k_pgen(const float* __restrict__ attn,
                       const float* __restrict__ dec,
                       const float* __restrict__ encW,
                       const float* __restrict__ wgen,
                       const float* __restrict__ bgen,
                       float* __restrict__ pgen) {
    int b    = blockIdx.y;
    int t0   = blockIdx.x * 16;
    int lane = threadIdx.x;            // 0..31, full wave, EXEC all ones
    int M    = lane & 15;              // row of A this lane carries
    int klo  = (lane >> 4) << 1;       // 0 for lanes 0-15, 2 for lanes 16-31

    const float* arow = attn + (size_t)(b * TT + t0 + M) * TS;
    const float* drow = dec  + (size_t)(b * TT + t0 + M) * HH;
    const float* ew   = encW + b * TS;
    const float* wg2  = wgen + HH;

    v8f c = {};

    // K over source positions: context part (attn . encW)
    for (int k0 = 0; k0 < TS; k0 += 4) {
        v2f a;
        a.x = arow[k0 + klo];
        a.y = arow[k0 + klo + 1];
        float e0 = ew[k0 + klo];
        float e1 = ew[k0 + klo + 1];
        v2f bm;
        bm.x = (M == 0) ? e0 : 0.f;    // only N==0 column is live
        bm.y = (M == 0) ? e1 : 0.f;
        c = __builtin_amdgcn_wmma_f32_16x16x4_f32(false, a, false, bm,
                                                  (short)0, c, false, false);
    }
    // K over hidden dim: decoder part (dec . W_gen[H:2H])
    for (int h0 = 0; h0 < HH; h0 += 4) {
        v2f a;
        a.x = drow[h0 + klo];
        a.y = drow[h0 + klo + 1];
        float e0 = wg2[h0 + klo];
        float e1 = wg2[h0 + klo + 1];
        v2f bm;
        bm.x = (M == 0) ? e0 : 0.f;
        bm.y = (M == 0) ? e1 : 0.f;
        c = __builtin_amdgcn_wmma_f32_16x16x4_f32(false, a, false, bm,
                                                  (short)0, c, false, false);
    }

    // Column N==0 lives in lane 0 (M=r) and lane 16 (M=8+r).
    if (M == 0) {
        int   mbase = (lane == 0) ? 0 : 8;
        float bg    = bgen[0];
        for (int r = 0; r < 8; ++r) {
            float logit = c[r] + bg;
            pgen[b * TT + t0 + mbase + r] = 1.f / (1.f + __expf(-logit));
        }
    }
}

// ---------------------------------------------------------------------------
// Kernel 3 (bandwidth-bound main pass): per (b,t) row of V=32100:
//   online softmax stats -> chunked LDS scatter of attention mass -> log mix.
// One 256-thread block per row; 32KB LDS scatter chunks (ds_add_f32 atomics).
// ---------------------------------------------------------------------------
__global__ void k_main(const float* __restrict__ logits,
                       const float* __restrict__ attn,
                       const int*   __restrict__ sids,
                       const float* __restrict__ pgen,
                       float* __restrict__ out) {
    __shared__ float s_copy[CHUNK];
    __shared__ float red_m[256];
    __shared__ float red_s[256];
    __shared__ float bc[2];

    int blk = blockIdx.x;              // 0..1023 == b*TT + t
    int b   = blk >> 8;
    int tid = threadIdx.x;

    const float* row = logits + (size_t)blk * VV;

    // Pass 1: online (max, sum-of-exp) over the row.
    float m = -INFINITY, s = 0.f;
    for (int v = tid; v < VV; v += 256) {
        float x = row[v];
        if (x > m) { s = s * __expf(m - x) + 1.f; m = x; }
        else       { s += __expf(x - m); }
    }
    red_m[tid] = m;
    red_s[tid] = s;
    __syncthreads();
    for (int off = 128; off > 0; off >>= 1) {
        if (tid < off) {
            float m1 = red_m[tid], s1 = red_s[tid];
            float m2 = red_m[tid + off], s2 = red_s[tid + off];
            float mm = fmaxf(m1, m2);
            red_m[tid] = mm;
            red_s[tid] = s1 * __expf(m1 - mm) + s2 * __expf(m2 - mm);
        }
        __syncthreads();
    }
    if (tid == 0) { bc[0] = red_m[0]; bc[1] = 1.f / red_s[0]; }
    __syncthreads();
    float Mx   = bc[0];
    float invS = bc[1];

    float p   = pgen[blk];
    float omp = 1.f - p;
    const int*   ids  = sids + b * TS;
    const float* arow = attn + (size_t)blk * TS;
    float*       orow = out  + (size_t)blk * VV;

    // Pass 2: chunked scatter + fused log output.
    for (int c0 = 0; c0 < VV; c0 += CHUNK) {
        int cend = (c0 + CHUNK < VV) ? (c0 + CHUNK) : VV;
        for (int i = tid; i < CHUNK; i += 256) s_copy[i] = 0.f;
        __syncthreads();
        for (int si = tid; si < TS; si += 256) {
            int id = ids[si];
            if (id >= c0 && id < cend)
                atomicAdd(&s_copy[id - c0], arow[si]);   // ds_add_f32
        }
        __syncthreads();
        for (int v = c0 + tid; v < cend; v += 256) {
            float x   = row[v];
            float val = p * __expf(x - Mx) * invS
                      + omp * s_copy[v - c0] + 1e-12f;
            orow[v] = __logf(val);
        }
        __syncthreads();
    }
}

// ---------------------------------------------------------------------------
extern "C" void kernel_launch(void* const* d_in, const int* in_sizes, int n_in,
                              void* d_out, int out_size, void* d_ws, size_t ws_size,
                              hipStream_t stream) {
    (void)in_sizes; (void)n_in; (void)out_size; (void)ws_size;
    const float* dec    = (const float*)d_in[0];   // (B,T,H)
    const float* attn   = (const float*)d_in[1];   // (B,T,S)
    const float* enc    = (const float*)d_in[2];   // (B,S,H)
    const float* logits = (const float*)d_in[3];   // (B,T,V)
    const float* wgen   = (const float*)d_in[4];   // (2H,1)
    const float* bgen   = (const float*)d_in[5];   // (1,)
    const int*   sids   = (const int*)d_in[6];     // (B,S)
    float* out  = (float*)d_out;

    float* encW = (float*)d_ws;                    // BB*TS floats
    float* pgen = encW + BB * TS;                  // BB*TT floats

    k_encw<<<(BB * TS) / 8, 256, 0, stream>>>(enc, wgen, encW);
    k_pgen<<<dim3(TT / 16, BB), 32, 0, stream>>>(attn, dec, encW, wgen, bgen, pgen);
    k_main<<<BB * TT, 256, 0, stream>>>(logits, attn, sids, pgen, out);
}